// ExcInference_24103356465642
// MI455X (gfx1250) — compile-verified
//
#include <hip/hip_runtime.h>
#include <hip/hip_bf16.h>

#define IDIM   80
#define HDIM   512
#define CDIM   64
#define P      80          // IN_EXTRA
#define WIN    160         // window = P + IDIM
#define NSHIFT 81          // P + 1
#define DDIM   160         // ODIM * OUT_EXTRA
#define KSEL   128         // 2 * CDIM
#define NPOS   2048        // B * T

typedef __attribute__((ext_vector_type(2))) float v2f;
typedef __attribute__((ext_vector_type(8))) float v8f;

__global__ __launch_bounds__(256) void exc_inference_kernel(
    const float* __restrict__ x,          // (2048, 80)
    const float* __restrict__ W_enc,      // (512, 160)
    const float* __restrict__ b_enc,      // (512)
    const float* __restrict__ W_dec,      // (160, 512)
    const float* __restrict__ b_dec,      // (160)
    const int*   __restrict__ mask_prev,  // (2048, 512)
    float* __restrict__ out)              // (2048, 80)
{
    __shared__ float         xpad[256];   // zero-padded window source (240 used)
    __shared__ float         energy[96];  // per-shift ||h||^2 (81 used)
    __shared__ float         hbuf[HDIM];  // h at best shift
    __shared__ unsigned int  ebits[HDIM]; // bit pattern of h*h (monotone for >=0)
    __shared__ unsigned char keep[HDIM];  // top-128 mask
    __shared__ float         xext[DDIM];  // decoder output
    __shared__ int           s_hind;
    __shared__ unsigned int  s_cnt;

    const int pos  = blockIdx.x;
    const int tid  = threadIdx.x;
    const int lane = tid & 31;
    const int wave = tid >> 5;
    const int half = lane >> 4;   // 0: lanes 0-15, 1: lanes 16-31
    const int l15  = lane & 15;

    // ---- init LDS ----
    xpad[tid] = 0.0f;
    if (tid < 96) energy[tid] = 0.0f;
    __syncthreads();
    if (tid < IDIM) xpad[P + tid] = x[pos * IDIM + tid];
    __syncthreads();

    // ---- Phase 1: encoder energies via f32 WMMA ----
    // C(96x512) = A(96x160) * B(160x512), A[m][k] = xpad[m+k], B[k][n] = W_enc[n*WIN+k]
    for (int tile = wave; tile < 6 * 32; tile += 8) {
        const int tm = tile >> 5;              // M tile 0..5
        const int tn = tile & 31;              // N tile 0..31
        const int mg = tm * 16 + l15;          // A row for this lane
        const int ng = tn * 16 + l15;          // B column for this lane
        const float* wrow = W_enc + ng * WIN;
        v8f c = {0.f, 0.f, 0.f, 0.f, 0.f, 0.f, 0.f, 0.f};
        for (int k = 0; k < WIN; k += 4) {
            const int kk = k + half * 2;
            v2f a, b;
            a.x = xpad[mg + kk];
            a.y = xpad[mg + kk + 1];
            b.x = wrow[kk];
            b.y = wrow[kk + 1];
            c = __builtin_amdgcn_wmma_f32_16x16x4_f32(
                    false, a, false, b, (short)0, c, false, false);
        }
        const float bias = b_enc[ng];
#pragma unroll
        for (int r = 0; r < 8; ++r) {
            const int m = tm * 16 + r + half * 8;   // shift index of this element
            float hv = c[r] + bias;
            float e  = hv * hv;
            // sum across the 16 lanes sharing this row (16 N values of the tile)
            e += __shfl_xor(e, 1, 16);
            e += __shfl_xor(e, 2, 16);
            e += __shfl_xor(e, 4, 16);
            e += __shfl_xor(e, 8, 16);
            if (l15 == 0 && m < NSHIFT) atomicAdd(&energy[m], e);
        }
    }
    __syncthreads();

    // ---- Phase 2: argmax over shifts (first max wins, like jnp.argmax) ----
    if (tid == 0) {
        float best = energy[0]; int bi = 0;
        for (int s = 1; s < NSHIFT; ++s) {
            float e = energy[s];
            if (e > best) { best = e; bi = s; }
        }
        s_hind = bi;
    }
    __syncthreads();
    const int hind = s_hind;

    // ---- Phase 3: recompute h at best shift, apply mask_prev ----
    for (int hh = tid; hh < HDIM; hh += 256) {
        const float* wrow = W_enc + hh * WIN;
        const float* xp   = xpad + hind;
        float acc = b_enc[hh];
        for (int k = 0; k < WIN; ++k) acc += wrow[k] * xp[k];
        if (mask_prev[pos * HDIM + hh] != 0) acc = 0.0f;
        hbuf[hh]  = acc;
        ebits[hh] = __float_as_uint(acc * acc);
    }
    __syncthreads();

    // ---- Phase 4: radix-select the 128th-largest squared value ----
    unsigned int v = 0u;
    for (int bit = 31; bit >= 0; --bit) {
        const unsigned int cand = v | (1u << bit);
        if (tid == 0) s_cnt = 0u;
        __syncthreads();
        unsigned int c = 0u;
        for (int hh = tid; hh < HDIM; hh += 256) c += (ebits[hh] >= cand) ? 1u : 0u;
        atomicAdd(&s_cnt, c);
        __syncthreads();
        if (s_cnt >= KSEL) v = cand;   // uniform across block
        __syncthreads();
    }
    // stable tie handling: strictly-greater always kept, equals filled in index order
    if (tid == 0) {
        int kept = 0;
        for (int i = 0; i < HDIM; ++i) {
            keep[i] = (ebits[i] > v) ? (unsigned char)1 : (unsigned char)0;
            kept += keep[i];
        }
        for (int i = 0; i < HDIM && kept < KSEL; ++i) {
            if (!keep[i] && ebits[i] == v) { keep[i] = 1; ++kept; }
        }
    }
    __syncthreads();
    for (int hh = tid; hh < HDIM; hh += 256) {
        if (!keep[hh]) hbuf[hh] = 0.0f;
    }
    __syncthreads();

    // ---- Phase 5: decoder matvec (160 x 512) ----
    if (tid < DDIM) {
        const float* wrow = W_dec + tid * HDIM;
        float acc = b_dec[tid];
        for (int hh = 0; hh < HDIM; ++hh) acc += wrow[hh] * hbuf[hh];
        xext[tid] = acc;
    }
    __syncthreads();

    // ---- Phase 6: gather 80-wide window at base = (D - idim) - h_ind ----
    if (tid < IDIM) {
        out[pos * IDIM + tid] = xext[(DDIM - IDIM) - hind + tid];
    }
}

extern "C" void kernel_launch(void* const* d_in, const int* in_sizes, int n_in,
                              void* d_out, int out_size, void* d_ws, size_t ws_size,
                              hipStream_t stream) {
    const float* x         = (const float*)d_in[0];
    const float* W_enc     = (const float*)d_in[1];
    const float* b_enc     = (const float*)d_in[2];
    const float* W_dec     = (const float*)d_in[3];
    const float* b_dec     = (const float*)d_in[4];
    const int*   mask_prev = (const int*)d_in[5];
    float* out = (float*)d_out;

    exc_inference_kernel<<<NPOS, 256, 0, stream>>>(
        x, W_enc, b_enc, W_dec, b_dec, mask_prev, out);
}